// StageModule_30202210025652
// MI455X (gfx1250) — compile-verified
//
#include <hip/hip_runtime.h>
#include <hip/hip_bf16.h>
#include <math.h>

typedef __attribute__((ext_vector_type(2))) float v2f;
typedef __attribute__((ext_vector_type(8))) float v8f;

#define B_    16
#define N_TOK 4096
#define NP1   4097
#define C_    384
#define NKEEP 2048           // pruned count N_
#define MROWS 2050           // 2049 keep rows + 1 merge token
#define MPAD  2080           // padded to 65 * 32 for tiling
#define KC    64             // K chunk staged in LDS
#define LDP   68             // LDS row pitch (floats): 16B-aligned, bank-friendly

// ---------------------------------------------------------------------------
// 1) Per-batch descending bitonic argsort of global_attn in LDS + wsum reduce
// ---------------------------------------------------------------------------
__global__ __launch_bounds__(1024)
void sort_kernel(const float* __restrict__ ga, int* __restrict__ order,
                 float* __restrict__ wsum) {
    __shared__ float keys[N_TOK];
    __shared__ int   vals[N_TOK];
    const int b = blockIdx.x;
    const float* g = ga + (size_t)b * N_TOK;
    for (int i = threadIdx.x; i < N_TOK; i += blockDim.x) { keys[i] = g[i]; vals[i] = i; }
    __syncthreads();
    for (int k = 2; k <= N_TOK; k <<= 1) {
        for (int j = k >> 1; j > 0; j >>= 1) {
            for (int i = threadIdx.x; i < N_TOK; i += blockDim.x) {
                int ixj = i ^ j;
                if (ixj > i) {
                    bool up = ((i & k) == 0);            // descending overall
                    float ki = keys[i], kj = keys[ixj];
                    bool sw = up ? (ki < kj) : (ki > kj);
                    if (sw) {
                        keys[i] = kj; keys[ixj] = ki;
                        int t = vals[i]; vals[i] = vals[ixj]; vals[ixj] = t;
                    }
                }
            }
            __syncthreads();
        }
    }
    float s = 0.f;
    for (int i = threadIdx.x; i < N_TOK; i += blockDim.x) {
        order[(size_t)b * N_TOK + i] = vals[i];
        if (i >= NKEEP) s += keys[i];                    // dropped-half weight sum
    }
    __syncthreads();
    keys[threadIdx.x] = s;
    __syncthreads();
    for (int st = 512; st > 0; st >>= 1) {
        if (threadIdx.x < (unsigned)st) keys[threadIdx.x] += keys[threadIdx.x + st];
        __syncthreads();
    }
    if (threadIdx.x == 0) wsum[b] = keys[0];
}

// ---------------------------------------------------------------------------
// 2) Gather CLS + top-2048 rows into Xcur; zero padding rows
// ---------------------------------------------------------------------------
__global__ __launch_bounds__(96)
void gather_kernel(const float* __restrict__ x_, const int* __restrict__ order,
                   float* __restrict__ Xcur) {
    const int r = blockIdx.x % MPAD;
    const int b = blockIdx.x / MPAD;
    const int t = threadIdx.x;                            // 96 * float4 = 384
    float4* dst = (float4*)(Xcur + ((size_t)b * MPAD + r) * C_);
    if (r == NKEEP + 1) return;                           // merge row written elsewhere
    if (r > NKEEP + 1) { dst[t] = make_float4(0.f, 0.f, 0.f, 0.f); return; }
    const float* src;
    if (r == 0) src = x_ + (size_t)b * NP1 * C_;          // CLS token
    else        src = x_ + ((size_t)b * NP1 + 1 + order[(size_t)b * N_TOK + (r - 1)]) * C_;
    dst[t] = ((const float4*)src)[t];
}

// ---------------------------------------------------------------------------
// 3) Weighted merge of dropped tokens -> Xcur row 2049; zero raw_total
// ---------------------------------------------------------------------------
__global__ __launch_bounds__(96)
void merge_kernel(const float* __restrict__ x_, const float* __restrict__ ga,
                  const int* __restrict__ order, const float* __restrict__ wsum,
                  float* __restrict__ Xcur, float* __restrict__ rawtotal) {
    const int b = blockIdx.y;
    const int c = blockIdx.x * 96 + threadIdx.x;
    const int* ord = order + (size_t)b * N_TOK;
    float acc = 0.f;
    for (int i = NKEEP; i < N_TOK; ++i) {
        int idx = ord[i];
        float w = ga[(size_t)b * N_TOK + idx];
        acc += w * x_[((size_t)b * NP1 + 1 + idx) * C_ + c];
    }
    Xcur[((size_t)b * MPAD + (NKEEP + 1)) * C_ + c] = acc / wsum[b];
    rawtotal[(size_t)b * C_ + c] = 0.f;
}

// ---------------------------------------------------------------------------
// 4) fp32 WMMA GEMM: 8 waves/WG, 32x64 tile, one 16x16 WMMA tile per wave.
//    MODE 0: Out = gelu(A@W + bias)            (exact erf GELU)
//    MODE 1: Out += A@W + bias ; row 2049 -> rawtotal
// ---------------------------------------------------------------------------
__device__ __forceinline__ float gelu_exact(float x) {
    return 0.5f * x * (1.f + erff(x * 0.70710678118654752f));
}

template <int MODE>
__global__ __launch_bounds__(256)
void gemm_kernel(const float* __restrict__ A, const float* __restrict__ W,
                 const float* __restrict__ bias, float* __restrict__ Out,
                 float* __restrict__ rawtotal) {
    __shared__ float As[32 * LDP];
    __shared__ float Bs[64 * LDP];
    const int b  = blockIdx.z;
    const int m0 = blockIdx.x * 32;
    const int n0 = blockIdx.y * 64;
    const float* Ab = A + (size_t)b * MPAD * C_;
    const int tid  = threadIdx.x;
    const int lane = tid & 31;
    const int w    = tid >> 5;
    const int wm   = w >> 2;          // 0..1  (16-row slab)
    const int wn   = w & 3;           // 0..3  (16-col slab)

    v8f acc;
    const float bv = bias[n0 + wn * 16 + (lane & 15)];
    #pragma unroll
    for (int r = 0; r < 8; ++r) acc[r] = bv;

    for (int k0 = 0; k0 < C_; k0 += KC) {
        // stage A chunk (32 x 64) with float4
        for (int i = tid; i < 32 * 16; i += 256) {
            int rr = i >> 4, cc = (i & 15) << 2;
            float4 v = *(const float4*)(Ab + (size_t)(m0 + rr) * C_ + k0 + cc);
            *(float4*)(&As[rr * LDP + cc]) = v;
        }
        // stage W chunk transposed: Bs[n][k] = W[(k0+kk)*C + n0+nn]
        for (int i = tid; i < 64 * 64; i += 256) {
            int kk = i >> 6, nn = i & 63;
            Bs[nn * LDP + kk] = W[(size_t)(k0 + kk) * C_ + n0 + nn];
        }
        __syncthreads();

        // A frag: lanes 0-15 -> M=lane, K={k,k+1}; lanes 16-31 -> K={k+2,k+3}
        const int ar = (wm * 16 + (lane & 15)) * LDP + ((lane >> 4) << 1);
        const int br = (wn * 16 + (lane & 15)) * LDP + ((lane >> 4) << 1);
        #pragma unroll
        for (int k = 0; k < KC; k += 4) {
            v2f av = *(const v2f*)(&As[ar + k]);
            v2f bw = *(const v2f*)(&Bs[br + k]);
            acc = __builtin_amdgcn_wmma_f32_16x16x4_f32(
                false, av, false, bw, (short)0, acc, false, false);
        }
        __syncthreads();
    }

    // C/D layout: v[r] lanes0-15 -> M=r, lanes16-31 -> M=r+8; N = lane&15
    const int nn    = n0 + wn * 16 + (lane & 15);
    const int mbase = m0 + wm * 16 + ((lane >> 4) << 3);
    if (MODE == 0) {
        #pragma unroll
        for (int r = 0; r < 8; ++r)
            Out[((size_t)b * MPAD + mbase + r) * C_ + nn] = gelu_exact(acc[r]);
    } else {
        #pragma unroll
        for (int r = 0; r < 8; ++r) {
            int m = mbase + r;
            float v = acc[r];
            size_t o = ((size_t)b * MPAD + m) * C_ + nn;
            Out[o] += v;                                        // residual
            if (m == NKEEP + 1) rawtotal[(size_t)b * C_ + nn] += v;
        }
    }
}

// ---------------------------------------------------------------------------
// 5) Final concat: [Xcur rows 0..2048 | x_drop + raw_total]
// ---------------------------------------------------------------------------
__global__ __launch_bounds__(96)
void final_kernel(const float* __restrict__ Xcur, const float* __restrict__ x_,
                  const int* __restrict__ order, const float* __restrict__ rawtotal,
                  float* __restrict__ out) {
    const int r = blockIdx.x % NP1;
    const int b = blockIdx.x / NP1;
    const int t = threadIdx.x;
    float4* dst = (float4*)(out + ((size_t)b * NP1 + r) * C_);
    if (r <= NKEEP) {
        dst[t] = ((const float4*)(Xcur + ((size_t)b * MPAD + r) * C_))[t];
    } else {
        int i   = r - (NKEEP + 1);
        int idx = order[(size_t)b * N_TOK + NKEEP + i];
        float4 xv = ((const float4*)(x_ + ((size_t)b * NP1 + 1 + idx) * C_))[t];
        float4 rv = ((const float4*)(rawtotal + (size_t)b * C_))[t];
        dst[t] = make_float4(xv.x + rv.x, xv.y + rv.y, xv.z + rv.z, xv.w + rv.w);
    }
}

// ---------------------------------------------------------------------------
extern "C" void kernel_launch(void* const* d_in, const int* in_sizes, int n_in,
                              void* d_out, int out_size, void* d_ws, size_t ws_size,
                              hipStream_t stream) {
    const float* x_ = (const float*)d_in[0];
    const float* ga = (const float*)d_in[1];
    // d_in[2] = ori_indices (unused by reference output)
    const float* W1 = (const float*)d_in[3];
    const float* b1 = (const float*)d_in[4];
    const float* W2 = (const float*)d_in[5];
    const float* b2 = (const float*)d_in[6];
    float* out = (float*)d_out;

    char* ws = (char*)d_ws;
    const size_t XB = (size_t)B_ * MPAD * C_ * sizeof(float);   // ~51.1 MB
    float* Xcur     = (float*)(ws);
    float* H        = (float*)(ws + XB);
    int*   order    = (int*)  (ws + 2 * XB);
    float* wsum     = (float*)(ws + 2 * XB + (size_t)B_ * N_TOK * sizeof(int));
    float* rawtotal = wsum + 16;                                 // 16B aligned

    sort_kernel<<<B_, 1024, 0, stream>>>(ga, order, wsum);
    gather_kernel<<<B_ * MPAD, 96, 0, stream>>>(x_, order, Xcur);
    merge_kernel<<<dim3(C_ / 96, B_), 96, 0, stream>>>(x_, ga, order, wsum, Xcur, rawtotal);

    dim3 ggrid(MPAD / 32, C_ / 64, B_);
    for (int l = 0; l < 2; ++l) {
        gemm_kernel<0><<<ggrid, 256, 0, stream>>>(Xcur, W1 + (size_t)l * C_ * C_,
                                                  b1 + (size_t)l * C_, H, nullptr);
        gemm_kernel<1><<<ggrid, 256, 0, stream>>>(H, W2 + (size_t)l * C_ * C_,
                                                  b2 + (size_t)l * C_, Xcur, rawtotal);
    }
    final_kernel<<<B_ * NP1, 96, 0, stream>>>(Xcur, x_, order, rawtotal, out);
}